// SocialModel2_46102178955997
// MI455X (gfx1250) — compile-verified
//
#include <hip/hip_runtime.h>
#include <cstdint>

// ---------------- problem constants (from reference) ----------------
#define USERN 100000
#define ITEMN 50000
#define TAGN  20000
#define DIMK  64
#define NNODE (USERN + ITEMN)   // 150000 interaction-graph nodes
#define MNODE (ITEMN + TAGN)    // 70000 tag-graph nodes
#define LEAKYF 0.5f
#define KEEPF  0.9f
#define E_ADJ 2000000
#define E_TAG 1000000
#define E_SOC 1000000

#define CHUNK 1024              // edges staged per block via TDM
#define TPB   256               // 8 wave32 waves

typedef unsigned int v4u __attribute__((ext_vector_type(4)));
typedef int          v8i __attribute__((ext_vector_type(8)));
typedef int          v4i __attribute__((ext_vector_type(4)));

// ---------------- device helpers ----------------
__device__ __forceinline__ float lrelu(float x) {
    return x >= 0.0f ? x : LEAKYF * x;
}

__device__ __forceinline__ float4 lrelu4(float4 x) {
    return make_float4(lrelu(x.x), lrelu(x.y), lrelu(x.z), lrelu(x.w));
}

// pcg-style hash -> deterministic per-edge dropout
__device__ __forceinline__ unsigned hash_u32(unsigned v) {
    v = v * 747796405u + 2891336453u;
    unsigned w = ((v >> ((v >> 28) + 4u)) ^ v) * 277803737u;
    return (w >> 22) ^ w;
}

#if __has_builtin(__builtin_amdgcn_tensor_load_to_lds) && __has_builtin(__builtin_amdgcn_s_wait_tensorcnt)
#define USE_TDM 1
#else
#define USE_TDM 0
#endif

#if USE_TDM
// One 1-D TDM transfer: elements [0, rem) of a 32-bit array at gptr are the
// "tensor"; the tile is CHUNK elements, so the tail chunk zero-fills OOB reads
// instead of touching memory.  Only group0 (addresses) differs per call; the
// shape descriptor g1 is built once by the caller.
__device__ __forceinline__ void tdm_load_1d(unsigned lds_off, const void* gptr,
                                            const v8i& g1) {
    unsigned long long ga = (unsigned long long)(uintptr_t)gptr;
    v4u g0;
    g0.x = 1u;                                   // count = 1 valid descriptor
    g0.y = lds_off;                              // lds_addr (bytes)
    g0.z = (unsigned)ga;                         // global_addr[31:0]
    g0.w = (unsigned)((ga >> 32) & 0x01FFFFFFu)  // global_addr[56:32]
         | (2u << 30);                           // type = 2 ("image")
    v4i gz4 = {0, 0, 0, 0};                      // groups 2/3 unused (<=2-D)
    v8i gz8 = {0, 0, 0, 0, 0, 0, 0, 0};          // extra descriptor words unused
    __builtin_amdgcn_tensor_load_to_lds(g0, g1, gz4, gz4, gz8, 0);
}

__device__ __forceinline__ v8i tdm_shape_1d(unsigned rem_elems) {
    v8i g1;
    g1[0] = (int)(2u << 16);                         // data_size = 4 bytes
    g1[1] = (int)((rem_elems & 0xFFFFu) << 16);      // tensor_dim0[15:0]
    g1[2] = (int)((rem_elems >> 16) | (1u << 16));   // tensor_dim0[31:16] | tensor_dim1=1
    g1[3] = (int)((unsigned)CHUNK << 16);            // tile_dim0 = CHUNK
    g1[4] = 0;                                       // tile_dim1/2 = 0 (unused)
    g1[5] = (int)rem_elems;                          // tensor_dim0_stride (unused, 1-D)
    g1[6] = 0;
    g1[7] = 0;
    return g1;
}
#endif

// ---------------- kernels ----------------

// lat = concat(u, i); out(acc) = lat      (float4 streaming)
__global__ __launch_bounds__(TPB) void init_lat_kernel(
    const float4* __restrict__ uE, const float4* __restrict__ iE,
    float4* __restrict__ lat, float4* __restrict__ out) {
    const int n4 = NNODE * (DIMK / 4);
    const int u4 = USERN * (DIMK / 4);
    for (int i = blockIdx.x * blockDim.x + threadIdx.x; i < n4;
         i += gridDim.x * blockDim.x) {
        float4 v = (i < u4) ? uE[i] : iE[i - u4];
        lat[i] = v;
        out[i] = v;
    }
}

// tag_lat = concat(i, t)
__global__ __launch_bounds__(TPB) void init_tag_kernel(
    const float4* __restrict__ iE, const float4* __restrict__ tE,
    float4* __restrict__ tagLat) {
    const int n4 = MNODE * (DIMK / 4);
    const int i4 = ITEMN * (DIMK / 4);
    for (int i = blockIdx.x * blockDim.x + threadIdx.x; i < n4;
         i += gridDim.x * blockDim.x) {
        tagLat[i] = (i < i4) ? iE[i] : tE[i - i4];
    }
}

// SpMM over an edge chunk: out[rows[e]] += drop(vals[e]) * src[cols[e]]
// src row = (c < splitCol) ? srcA[c] : srcB[c].
// One block == one CHUNK of edges, staged in LDS by the Tensor Data Mover.
// 16 lanes cooperate per edge: one float4 (16 B) each -> coalesced 256 B row.
__global__ __launch_bounds__(TPB) void spmm_kernel(
    const int* __restrict__ rows, const int* __restrict__ cols,
    const float* __restrict__ vals,
    const float* __restrict__ srcA, const float* __restrict__ srcB,
    int splitCol, float* __restrict__ outAcc, int nEdges, unsigned seed) {
    __shared__ unsigned ebuf[3 * CHUNK];  // [rows | cols | vals] 12 KB

    const int start = (int)blockIdx.x * CHUNK;
    if (start >= nEdges) return;
    const int count = min(CHUNK, nEdges - start);

#if USE_TDM
    if (threadIdx.x < 32) {  // wave 0 only issues the three DMAs
        const unsigned base = (unsigned)(uintptr_t)(&ebuf[0]);  // low 32b = LDS offset
        const v8i shape = tdm_shape_1d((unsigned)(nEdges - start));
        tdm_load_1d(base,                   rows + start, shape);
        tdm_load_1d(base + CHUNK * 4u,      cols + start, shape);
        tdm_load_1d(base + 2u * CHUNK * 4u, vals + start, shape);
        __builtin_amdgcn_s_wait_tensorcnt(0);
    }
    __syncthreads();
#else
    for (int i = threadIdx.x; i < count; i += TPB) {
        ebuf[i]             = (unsigned)rows[start + i];
        ebuf[CHUNK + i]     = (unsigned)cols[start + i];
        ebuf[2 * CHUNK + i] = __float_as_uint(vals[start + i]);
    }
    __syncthreads();
#endif

    const int lane = threadIdx.x & 15;           // 16 lanes per edge
    for (int e = (int)(threadIdx.x >> 4); e < count; e += (TPB >> 4)) {
        const int   r = (int)ebuf[e];
        const int   c = (int)ebuf[CHUNK + e];
        float       v = __uint_as_float(ebuf[2 * CHUNK + e]);

        // SpAdjDropEdge: keep w.p. KEEP, rescale by 1/KEEP (deterministic hash)
        unsigned h   = hash_u32((unsigned)(start + e) ^ seed);
        float    u01 = (float)(h >> 8) * (1.0f / 16777216.0f);
        v = (u01 + KEEPF >= 1.0f) ? v * (1.0f / KEEPF) : 0.0f;

        if (v != 0.0f) {
            const float* src =
                (c < splitCol) ? (srcA + (size_t)c * DIMK)
                               : (srcB + (size_t)c * DIMK);
            const float4 x =
                *reinterpret_cast<const float4*>(src + lane * 4);
            float* o = outAcc + (size_t)r * DIMK + lane * 4;
            unsafeAtomicAdd(o + 0, v * x.x);   // global_atomic_add_f32 in L2
            unsafeAtomicAdd(o + 1, v * x.y);
            unsafeAtomicAdd(o + 2, v * x.z);
            unsafeAtomicAdd(o + 3, v * x.w);
        }
    }
}

// users: lat = lrelu(tem_u) + lrelu(soc); acc += lat
__global__ __launch_bounds__(TPB) void combine_user_kernel(
    const float4* __restrict__ tem, const float4* __restrict__ soc,
    float4* __restrict__ lat, float4* __restrict__ out) {
    const int n4 = USERN * (DIMK / 4);
    for (int i = blockIdx.x * blockDim.x + threadIdx.x; i < n4;
         i += gridDim.x * blockDim.x) {
        float4 a = lrelu4(tem[i]);
        float4 s = lrelu4(soc[i]);
        float4 l = make_float4(a.x + s.x, a.y + s.y, a.z + s.z, a.w + s.w);
        lat[i] = l;
        float4 o = out[i];
        out[i] = make_float4(o.x + l.x, o.y + l.y, o.z + l.z, o.w + l.w);
    }
}

// items: lat_i = lrelu(tem_i) + lrelu(tg[:ITEM]); acc += lat_i
__global__ __launch_bounds__(TPB) void combine_item_kernel(
    const float4* __restrict__ temI, const float4* __restrict__ tg,
    float4* __restrict__ latI, float4* __restrict__ outI) {
    const int n4 = ITEMN * (DIMK / 4);
    for (int i = blockIdx.x * blockDim.x + threadIdx.x; i < n4;
         i += gridDim.x * blockDim.x) {
        float4 a = lrelu4(temI[i]);
        float4 g = lrelu4(tg[i]);
        float4 l = make_float4(a.x + g.x, a.y + g.y, a.z + g.z, a.w + g.w);
        latI[i] = l;
        float4 o = outI[i];
        outI[i] = make_float4(o.x + l.x, o.y + l.y, o.z + l.z, o.w + l.w);
    }
}

// tag_lat = lrelu(tg)   (full M rows)
__global__ __launch_bounds__(TPB) void tag_update_kernel(
    const float4* __restrict__ tg, float4* __restrict__ tagLat) {
    const int n4 = MNODE * (DIMK / 4);
    for (int i = blockIdx.x * blockDim.x + threadIdx.x; i < n4;
         i += gridDim.x * blockDim.x) {
        tagLat[i] = lrelu4(tg[i]);
    }
}

// ---------------- launcher ----------------
extern "C" void kernel_launch(void* const* d_in, const int* in_sizes, int n_in,
                              void* d_out, int out_size, void* d_ws,
                              size_t ws_size, hipStream_t stream) {
    const float* uE       = (const float*)d_in[0];
    const float* iE       = (const float*)d_in[1];
    const float* tE       = (const float*)d_in[2];
    const int*   adj_rows = (const int*)d_in[3];
    const int*   adj_cols = (const int*)d_in[4];
    const float* adj_vals = (const float*)d_in[5];
    const int*   tag_rows = (const int*)d_in[6];
    const int*   tag_cols = (const int*)d_in[7];
    const float* tag_vals = (const float*)d_in[8];
    const int*   soc_rows = (const int*)d_in[9];
    const int*   soc_cols = (const int*)d_in[10];
    const float* soc_vals = (const float*)d_in[11];

    float* out = (float*)d_out;  // acc, [NNODE, 64]

    // workspace layout (floats)
    float* ws     = (float*)d_ws;
    float* lat    = ws;                            // NNODE*64
    float* tagLat = lat + (size_t)NNODE * DIMK;    // MNODE*64
    float* tem    = tagLat + (size_t)MNODE * DIMK; // NNODE*64
    float* tg     = tem + (size_t)NNODE * DIMK;    // MNODE*64
    float* soc    = tg + (size_t)MNODE * DIMK;     // USERN*64

    const int gN = (NNODE * (DIMK / 4) + TPB - 1) / TPB;
    const int gM = (MNODE * (DIMK / 4) + TPB - 1) / TPB;
    const int gU = (USERN * (DIMK / 4) + TPB - 1) / TPB;
    const int gI = (ITEMN * (DIMK / 4) + TPB - 1) / TPB;

    init_lat_kernel<<<gN, TPB, 0, stream>>>((const float4*)uE,
                                            (const float4*)iE, (float4*)lat,
                                            (float4*)out);
    init_tag_kernel<<<gM, TPB, 0, stream>>>((const float4*)iE,
                                            (const float4*)tE,
                                            (float4*)tagLat);

    const int bAdj = (E_ADJ + CHUNK - 1) / CHUNK;
    const int bTag = (E_TAG + CHUNK - 1) / CHUNK;
    const int bSoc = (E_SOC + CHUNK - 1) / CHUNK;

    for (int layer = 0; layer < 2; ++layer) {
        (void)hipMemsetAsync(tem, 0, (size_t)NNODE * DIMK * sizeof(float), stream);
        (void)hipMemsetAsync(tg,  0, (size_t)MNODE * DIMK * sizeof(float), stream);
        (void)hipMemsetAsync(soc, 0, (size_t)USERN * DIMK * sizeof(float), stream);

        const unsigned s0 = (unsigned)(3 * layer + 0) * 2654435761u + 12345u;
        const unsigned s1 = (unsigned)(3 * layer + 1) * 2654435761u + 12345u;
        const unsigned s2 = (unsigned)(3 * layer + 2) * 2654435761u + 12345u;

        // tem = gcn(adj, lat)          cols index lat directly (split=NNODE)
        spmm_kernel<<<bAdj, TPB, 0, stream>>>(adj_rows, adj_cols, adj_vals,
                                              lat, lat, NNODE, tem, E_ADJ, s0);
        // tg = gcn(tag, [lat[USER:], tagLat[ITEM:]]):
        //   c < ITEM -> lat[USER + c] ; c >= ITEM -> tagLat[c]
        spmm_kernel<<<bTag, TPB, 0, stream>>>(tag_rows, tag_cols, tag_vals,
                                              lat + (size_t)USERN * DIMK,
                                              tagLat, ITEMN, tg, E_TAG, s1);
        // soc = gcn(soc, lat[:USER])   cols < USER index lat directly
        spmm_kernel<<<bSoc, TPB, 0, stream>>>(soc_rows, soc_cols, soc_vals,
                                              lat, lat, NNODE, soc, E_SOC, s2);

        combine_user_kernel<<<gU, TPB, 0, stream>>>(
            (const float4*)tem, (const float4*)soc, (float4*)lat,
            (float4*)out);
        combine_item_kernel<<<gI, TPB, 0, stream>>>(
            (const float4*)(tem + (size_t)USERN * DIMK), (const float4*)tg,
            (float4*)(lat + (size_t)USERN * DIMK),
            (float4*)(out + (size_t)USERN * DIMK));
        tag_update_kernel<<<gM, TPB, 0, stream>>>((const float4*)tg,
                                                  (float4*)tagLat);
    }
}